// VectorQuantizer_81664508166539
// MI455X (gfx1250) — compile-verified
//
#include <hip/hip_runtime.h>
#include <stdint.h>

typedef float v2f __attribute__((ext_vector_type(2)));
typedef float v4f __attribute__((ext_vector_type(4)));
typedef float v8f __attribute__((ext_vector_type(8)));

#define N_TOK 16384   // 16 * 32 * 32 tokens
#define NE    16384   // codebook entries
#define EDIM  64
#define BETA  0.25f

// output layout (floats, concatenated in reference return order)
#define OUT_ZQ    0           // 1048576 elements (16,64,32,32)
#define OUT_LOSS  1048576
#define OUT_PERP  1048577
#define OUT_USE   1048578
#define OUT_IDX   1048579     // 16384 elements

// workspace layout (bytes)
#define WS_ZF      0u          // float[N_TOK*EDIM]  = 4 MiB
#define WS_ENORM   4194304u    // float[NE]          = 64 KiB
#define WS_MINPACK 4259840u    // u64[N_TOK]         = 128 KiB (8B aligned)
#define WS_COUNTS  4390912u    // u32[NE]            = 64 KiB
#define WS_PART    4456448u    // float[64] partial MSE sums

#define CHUNK  32              // codebook entries staged per pipeline step
#define LDSPAD 68              // LDS row stride (floats): 4*l15+2*hi banks, conflict-free

// sched_group_barrier masks
#define SGB_WMMA   0x008
#define SGB_VMEMRD 0x020
#define SGB_DSRD   0x100

// order-preserving float -> u32 key (for packed atomicMin argmin)
__device__ __forceinline__ unsigned int fkey(float f) {
    unsigned int u = __float_as_uint(f);
    return (u & 0x80000000u) ? ~u : (u | 0x80000000u);
}

// ---------------------------------------------------------------------------
// 1) transpose z (b,c,h,w) -> zf[token][c], token = b*1024 + h*32 + w
__global__ __launch_bounds__(256) void vq_transpose(const float* __restrict__ z,
                                                    float* __restrict__ zf) {
    const int i  = blockIdx.x * 256 + threadIdx.x;   // linear over z (coalesced read)
    const int b  = i >> 16;
    const int c  = (i >> 10) & 63;
    const int hw = i & 1023;
    zf[(((b << 10) | hw) * EDIM) + c] = z[i];
}

// ---------------------------------------------------------------------------
// 2) enorm[n] = ||emb[n]||^2 ; init minpack / counts (ws is poisoned)
__global__ __launch_bounds__(256) void vq_prep(const float* __restrict__ emb,
                                               float* __restrict__ enorm,
                                               unsigned long long* __restrict__ minpack,
                                               unsigned int* __restrict__ counts) {
    const int t = blockIdx.x * 256 + threadIdx.x;    // 0..16383
    const v4f* er = (const v4f*)(emb + (size_t)t * EDIM);
    float s = 0.f;
#pragma unroll
    for (int i = 0; i < EDIM / 4; ++i) {
        v4f v = er[i];
        s = fmaf(v.x, v.x, s); s = fmaf(v.y, v.y, s);
        s = fmaf(v.z, v.z, s); s = fmaf(v.w, v.w, s);
    }
    enorm[t]   = s;
    minpack[t] = 0xFFFFFFFFFFFFFFFFull;
    counts[t]  = 0u;
}

// ---------------------------------------------------------------------------
// 3) WMMA argmin GEMM with LDS-staged, double-buffered B tiles and an
//    explicitly enforced software-pipelined schedule.
//    Each wave owns a 16-row A tile (pre-scaled by -2) per the 16x4 f32 A
//    layout; the block stages 32 codebook rows per chunk in LDS (row stride
//    68 floats -> ds_load fragments hit all 64 banks exactly once per wave).
//    A 17th k-step with a constant-1 A column adds enorm[n], so
//    C = ||e||^2 - 2 z.e directly.  sched_group_barriers keep 6
//    ds_load_2addr_b64 in flight ahead of the WMMA chain; launch_bounds
//    (256,2) relaxes the RA occupancy target so both accumulation chains get
//    rotating fragment buffers (avoids the skew-1 register-reuse collapse).
__global__ __launch_bounds__(256, 2) void vq_argmin_wmma(
    const float* __restrict__ zf, const float* __restrict__ emb,
    const float* __restrict__ enorm, unsigned long long* __restrict__ minpack)
{
    __shared__ float ldsB[2][CHUNK][LDSPAD];   // 2 x 8.5 KiB of the 320 KiB/WGP

    const int tid  = threadIdx.x;
    const int lane = tid & 31;
    const int wave = tid >> 5;
    const int m0   = (blockIdx.x * 8 + wave) * 16;
    const int l15  = lane & 15;
    const int hi   = lane >> 4;
    const int mrow = m0 + l15;

    // A tile preload, scaled by -2 (exact: power of two)
    v2f a[16];
#pragma unroll
    for (int s = 0; s < 16; ++s) {
        const int k = 4 * s + 2 * hi;
        a[s].x = -2.0f * zf[mrow * EDIM + k];
        a[s].y = -2.0f * zf[mrow * EDIM + k + 1];
    }
    v2f a17; a17.x = hi ? 0.f : 1.f; a17.y = 0.f;   // constant-1 K column

    float minv[8];
    int   mini[8];
#pragma unroll
    for (int r = 0; r < 8; ++r) { minv[r] = 3.4e38f; mini[r] = 0; }

    const int nbase0  = blockIdx.y * (NE / 4);      // 4 grid-y codebook slices
    const int nchunks = (NE / 4) / CHUNK;           // 128 chunks of 32 entries

    // cooperative fetch mapping: 8 threads per codebook row, 8 floats each
    const int frow = tid >> 3;                      // 0..31
    const int fcol = (tid & 7) * 8;                 // 0..56

    // prologue: stage chunk 0 into buffer 0
    {
        const float* src = emb + (size_t)(nbase0 + frow) * EDIM + fcol;
        v4f f0 = *(const v4f*)src;
        v4f f1 = *(const v4f*)(src + 4);
        *(v4f*)&ldsB[0][frow][fcol]     = f0;
        *(v4f*)&ldsB[0][frow][fcol + 4] = f1;
    }
    __syncthreads();

    for (int chunk = 0; chunk < nchunks; ++chunk) {
        const int p  = chunk & 1;
        const int nb = nbase0 + chunk * CHUNK;

        // branchless next-tile fetch: last iteration re-reads the slice base
        // (stored into the dead buffer, never consumed) to keep one scheduling
        // region -- no control flow inside the hot body.
        const int fbase = (chunk + 1 < nchunks) ? (nb + CHUNK) : nbase0;
        const float* src = emb + (size_t)(fbase + frow) * EDIM + fcol;
        v4f f0 = *(const v4f*)src;
        v4f f1 = *(const v4f*)(src + 4);

        const float en0 = enorm[nb + l15];
        const float en1 = enorm[nb + 16 + l15];

        // B fragments from LDS (backend pairs these into ds_load_2addr_b64)
        v2f b0[16], b1[16];
#pragma unroll
        for (int s = 0; s < 16; ++s)
            b0[s] = *(const v2f*)&ldsB[p][l15][4 * s + 2 * hi];
#pragma unroll
        for (int s = 0; s < 16; ++s)
            b1[s] = *(const v2f*)&ldsB[p][16 + l15][4 * s + 2 * hi];

        v8f c0 = {}, c1 = {};
#pragma unroll
        for (int s = 0; s < 16; ++s) {
            c0 = __builtin_amdgcn_wmma_f32_16x16x4_f32(
                     false, a[s], false, b0[s], (short)0, c0, false, false);
            c1 = __builtin_amdgcn_wmma_f32_16x16x4_f32(
                     false, a[s], false, b1[s], (short)0, c1, false, false);
        }
        {   // 17th k-step: + 1 * enorm[n]  ->  C = ||e||^2 - 2 z.e
            v2f t0; t0.x = hi ? 0.f : en0; t0.y = 0.f;
            v2f t1; t1.x = hi ? 0.f : en1; t1.y = 0.f;
            c0 = __builtin_amdgcn_wmma_f32_16x16x4_f32(
                     false, a17, false, t0, (short)0, c0, false, false);
            c1 = __builtin_amdgcn_wmma_f32_16x16x4_f32(
                     false, a17, false, t1, (short)0, c1, false, false);
        }

        // branch-free running argmin (EXEC stays all-ones for the WMMAs)
        const int col0 = nb + l15;
        const int col1 = nb + 16 + l15;
#pragma unroll
        for (int r = 0; r < 8; ++r) {
            const bool lt0 = c0[r] < minv[r];
            mini[r] = lt0 ? col0  : mini[r];
            minv[r] = lt0 ? c0[r] : minv[r];
            const bool lt1 = c1[r] < minv[r];
            mini[r] = lt1 ? col1  : mini[r];
            minv[r] = lt1 ? c1[r] : minv[r];
        }

        // ---- enforced schedule for this region ----
        // VMEM reads first (waits land late), then a 6-deep DS-read skew,
        // steady state {2 WMMA, 1 DS read}, tail WMMAs last.
        // Totals per chunk: 4 VMEM reads, 16 DS reads, 34 WMMAs.
        __builtin_amdgcn_sched_group_barrier(SGB_VMEMRD, 4, 0);
        __builtin_amdgcn_sched_group_barrier(SGB_DSRD,   6, 0);
#pragma unroll
        for (int g = 0; g < 10; ++g) {
            __builtin_amdgcn_sched_group_barrier(SGB_WMMA, 2, 0);
            __builtin_amdgcn_sched_group_barrier(SGB_DSRD, 1, 0);
        }
        __builtin_amdgcn_sched_group_barrier(SGB_WMMA, 14, 0);

        // pipeline: all waves done reading buffer p^1 two barriers ago
        __syncthreads();
        *(v4f*)&ldsB[p ^ 1][frow][fcol]     = f0;
        *(v4f*)&ldsB[p ^ 1][frow][fcol + 4] = f1;
        __syncthreads();
    }

    // reduce (min,idx) across the 16 lanes of each half-wave
#pragma unroll
    for (int off = 8; off >= 1; off >>= 1) {
#pragma unroll
        for (int r = 0; r < 8; ++r) {
            const float ov = __shfl_xor(minv[r], off, 32);
            const int   oi = __shfl_xor(mini[r], off, 32);
            const bool take = (ov < minv[r]) || (ov == minv[r] && oi < mini[r]);
            minv[r] = take ? ov : minv[r];
            mini[r] = take ? oi : mini[r];
        }
    }

    if (l15 == 0) {   // lanes 0 (rows m0..m0+7) and 16 (rows m0+8..m0+15)
#pragma unroll
        for (int r = 0; r < 8; ++r) {
            const int row = m0 + r + 8 * hi;
            const unsigned long long pack =
                ((unsigned long long)fkey(minv[r]) << 32) | (unsigned int)mini[r];
            atomicMin(&minpack[row], pack);
        }
    }
}

// ---------------------------------------------------------------------------
// 4) gather z_q, write z_q_out (b,c,h,w), idx floats, histogram, MSE partials
__global__ __launch_bounds__(256) void vq_gather(
    const float* __restrict__ zf, const float* __restrict__ emb,
    const unsigned long long* __restrict__ minpack,
    unsigned int* __restrict__ counts, float* __restrict__ partial,
    float* __restrict__ out)
{
    __shared__ float red[256];
    const int t = blockIdx.x * 256 + threadIdx.x;        // token
    const unsigned long long p = minpack[t];
    const int idx = (int)(unsigned int)(p & 0xFFFFFFFFull);
    atomicAdd(&counts[idx], 1u);
    out[OUT_IDX + t] = (float)idx;

    const int b  = t >> 10;
    const int hw = t & 1023;
    const float* er = emb + (size_t)idx * EDIM;
    const float* zr = zf  + (size_t)t   * EDIM;
    float ss = 0.f;
#pragma unroll 8
    for (int c = 0; c < EDIM; ++c) {
        const float q = er[c];
        const float d = q - zr[c];
        ss = fmaf(d, d, ss);
        out[OUT_ZQ + (((b * EDIM + c) << 10) | hw)] = q;  // straight-through == z_q
    }

    red[threadIdx.x] = ss;
    __syncthreads();
    for (int s = 128; s > 0; s >>= 1) {
        if (threadIdx.x < s) red[threadIdx.x] += red[threadIdx.x + s];
        __syncthreads();
    }
    if (threadIdx.x == 0) partial[blockIdx.x] = red[0];   // fixed slot: deterministic
}

// ---------------------------------------------------------------------------
// 5) finalize scalars: loss = 1.25 * MSE, perplexity, cluster_use
__global__ __launch_bounds__(256) void vq_finalize(
    const unsigned int* __restrict__ counts,
    const float* __restrict__ partial, float* __restrict__ out)
{
    __shared__ float redE[256];
    __shared__ float redU[256];
    float ent = 0.f, used = 0.f;
    for (int t = threadIdx.x; t < NE; t += 256) {
        const unsigned int c = counts[t];
        const float pr = (float)c * (1.0f / (float)N_TOK);
        ent  += pr * logf(pr + 1e-10f);
        used += (c > 0u) ? 1.f : 0.f;
    }
    redE[threadIdx.x] = ent;
    redU[threadIdx.x] = used;
    __syncthreads();
    for (int s = 128; s > 0; s >>= 1) {
        if (threadIdx.x < s) {
            redE[threadIdx.x] += redE[threadIdx.x + s];
            redU[threadIdx.x] += redU[threadIdx.x + s];
        }
        __syncthreads();
    }
    if (threadIdx.x == 0) {
        float ssum = 0.f;
        for (int i = 0; i < 64; ++i) ssum += partial[i];   // fixed order
        const float mse = ssum * (1.0f / 1048576.0f);
        out[OUT_LOSS] = mse * (1.0f + BETA);
        out[OUT_PERP] = expf(-redE[0]);
        out[OUT_USE]  = redU[0];
    }
}

// ---------------------------------------------------------------------------
extern "C" void kernel_launch(void* const* d_in, const int* in_sizes, int n_in,
                              void* d_out, int out_size, void* d_ws, size_t ws_size,
                              hipStream_t stream)
{
    const float* z   = (const float*)d_in[0];   // (16,64,32,32)
    const float* emb = (const float*)d_in[1];   // (16384,64)
    float* out = (float*)d_out;
    char*  ws  = (char*)d_ws;

    float*              zf      = (float*)(ws + WS_ZF);
    float*              enorm   = (float*)(ws + WS_ENORM);
    unsigned long long* minpack = (unsigned long long*)(ws + WS_MINPACK);
    unsigned int*       counts  = (unsigned int*)(ws + WS_COUNTS);
    float*              partial = (float*)(ws + WS_PART);

    vq_transpose<<<4096, 256, 0, stream>>>(z, zf);
    vq_prep<<<64, 256, 0, stream>>>(emb, enorm, minpack, counts);

    dim3 grid(128, 4);   // 128 M-blocks (8 waves x 16 rows) x 4 codebook slices
    vq_argmin_wmma<<<grid, 256, 0, stream>>>(zf, emb, enorm, minpack);

    vq_gather<<<64, 256, 0, stream>>>(zf, emb, minpack, counts, partial, out);
    vq_finalize<<<1, 256, 0, stream>>>(counts, partial, out);
}